// BlocCircLinear_67680094651012
// MI455X (gfx1250) — compile-verified
//
#include <hip/hip_runtime.h>

// Block-circulant linear layer for MI455X (gfx1250, wave32).
// y[b, m*16+p] = sum_{n,q} blocks[(m-n)%512][p][q] * x[b, n*16+q]
// V_WMMA_F32_16X16X4_F32 (exact f32 path). Each wave computes two adjacent
// output blocks (m0, m0+1) and slides a window over the circulant diagonal:
//   per 2-n step the blocks needed are {K+1, K, K, K-1}; K serves two tiles
//   and K-1 is carried into the next step as its K+1. -> 16 loads : 16 WMMAs.

typedef __attribute__((ext_vector_type(2))) float v2f;
typedef __attribute__((ext_vector_type(8))) float v8f;

#define BATCH 32
#define DIM   8192
#define NB    512      // number of circulant blocks
#define BS    16       // block size (p == q == 16)

static __device__ __forceinline__ v8f wmma_f32(v2f a, v2f b, v8f c) {
    return __builtin_amdgcn_wmma_f32_16x16x4_f32(
        false, a, false, b, (short)0, c, false, false);
}

// One workgroup = 64 threads = 2 waves; wave w handles batch half w.
// blockIdx.x -> output block pair (m0 = 2*bx, m1 = m0+1), blockIdx.y -> K-split.
__global__ void __launch_bounds__(64)
bcl_wmma_f32(const float* __restrict__ x,
             const float* __restrict__ blocks,
             float* __restrict__ out,   // [nsplit][BATCH][DIM] partials
             int nper)                  // n-blocks per split (even, >= 2)
{
    const int m0    = blockIdx.x * 2;
    const int split = blockIdx.y;
    const int n0    = split * nper;

    const int lane  = threadIdx.x & 31;
    const int waveb = threadIdx.x >> 5;   // batch half: rows [16w, 16w+16)
    const int half  = lane >> 4;          // lane half selects K pair / M group
    const int p     = lane & 15;          // output column within block (also A row)

    // A-matrix (16x4 of x): lane L -> row M = L%16, K = 2*half + {0,1}
    const float* xrow = x + (size_t)(waveb * 16 + p) * DIM + 2 * half;
    // B-matrix (4x16 = W^T): lane L -> N = p, K = 2*half + {0,1}
    //   B[K][N] = W[p][q], q = 4c + 2*half + j -> adjacent pair, b64 load
    const float* brow = blocks + (size_t)p * BS + 2 * half;

    v8f acc00 = {};  // (m0, n even chain)
    v8f acc01 = {};  // (m0, n odd  chain)
    v8f acc10 = {};  // (m1, n even chain)
    v8f acc11 = {};  // (m1, n odd  chain)

    // Prologue: carried fragment bh = blocks[(m1 - n0) % NB] = K+1 at first step.
    v2f bh[4];
    {
        const float* wh = brow + (((m0 + 1 - n0) & (NB - 1)) * (BS * BS));
#pragma unroll
        for (int c = 0; c < 4; ++c)
            bh[c] = *(const v2f*)(wh + 4 * c);
    }

    for (int n = n0; n < n0 + nper; n += 2) {
        const int K  = (m0 - n) & (NB - 1);   // block for (m0,n) and (m1,n+1)
        const int Kl = (K - 1)  & (NB - 1);   // block for (m0,n+1); next step's K+1

        const float* xa0 = xrow + n * BS;
        const float* xa1 = xa0 + BS;
        const float* wm  = brow + K  * (BS * BS);
        const float* wl  = brow + Kl * (BS * BS);

        v2f a0[4], a1[4], bm[4], bnext[4];
#pragma unroll
        for (int c = 0; c < 4; ++c) {
            a0[c]    = *(const v2f*)(xa0 + 4 * c);
            a1[c]    = *(const v2f*)(xa1 + 4 * c);
            bm[c]    = *(const v2f*)(wm  + 4 * c);
            bnext[c] = *(const v2f*)(wl  + 4 * c);
        }
        // 4 independent accumulation chains; bm feeds two of them.
#pragma unroll
        for (int c = 0; c < 4; ++c) {
            acc00 = wmma_f32(a0[c], bm[c],    acc00);   // (m0, n)
            acc10 = wmma_f32(a0[c], bh[c],    acc10);   // (m1, n)
            acc01 = wmma_f32(a1[c], bnext[c], acc01);   // (m0, n+1)
            acc11 = wmma_f32(a1[c], bm[c],    acc11);   // (m1, n+1)
        }
        // Slide window: this step's K-1 is next step's K+1 (loop phi, no copies).
#pragma unroll
        for (int c = 0; c < 4; ++c)
            bh[c] = bnext[c];
    }

    const v8f accM0 = acc00 + acc01;
    const v8f accM1 = acc10 + acc11;

    // D layout: VGPR v, lane L -> D[M = v + 8*half][N = p]
    float* ob0 = out + (size_t)split * (BATCH * DIM)
                     + (size_t)(waveb * 16 + 8 * half) * DIM
                     + m0 * BS + p;
    float* ob1 = ob0 + BS;   // m1 = m0 + 1
#pragma unroll
    for (int v = 0; v < 8; ++v) {
        ob0[(size_t)v * DIM] = accM0[v];
        ob1[(size_t)v * DIM] = accM1[v];
    }
}

// Deterministic reduction of the K-split partials.
__global__ void __launch_bounds__(256)
bcl_reduce(const float* __restrict__ part, float* __restrict__ out, int nsplit)
{
    const int i = blockIdx.x * blockDim.x + threadIdx.x;
    if (i < BATCH * DIM) {
        float s = 0.f;
        for (int k = 0; k < nsplit; ++k)
            s += part[(size_t)k * (BATCH * DIM) + i];
        out[i] = s;
    }
}

extern "C" void kernel_launch(void* const* d_in, const int* in_sizes, int n_in,
                              void* d_out, int out_size, void* d_ws, size_t ws_size,
                              hipStream_t stream) {
    const float* x      = (const float*)d_in[0];   // [32, 8192] f32
    const float* blocks = (const float*)d_in[1];   // [512, 16, 16] f32
    float*       out    = (float*)d_out;           // [32, 8192] f32

    const size_t OUT_ELEMS = (size_t)BATCH * DIM;

    // K-split for occupancy (512 tile-waves alone is too few); deterministic,
    // size-based choice so every call does identical work.
    int nsplit = 1;
    if      (ws_size >= 8 * OUT_ELEMS * sizeof(float)) nsplit = 8;
    else if (ws_size >= 4 * OUT_ELEMS * sizeof(float)) nsplit = 4;

    if (nsplit == 1) {
        dim3 grid(NB / 2, 1);
        bcl_wmma_f32<<<grid, 64, 0, stream>>>(x, blocks, out, NB);
    } else {
        float* part = (float*)d_ws;
        dim3 grid(NB / 2, nsplit);
        bcl_wmma_f32<<<grid, 64, 0, stream>>>(x, blocks, part, NB / nsplit);
        const int threads = 256;
        const int nblk = (int)((OUT_ELEMS + threads - 1) / threads);
        bcl_reduce<<<nblk, threads, 0, stream>>>(part, out, nsplit);
    }
}